// GATLayer_34239479284303
// MI455X (gfx1250) — compile-verified
//
#include <hip/hip_runtime.h>
#include <hip/hip_bf16.h>
#include <math.h>

typedef __attribute__((ext_vector_type(2))) float    v2f;
typedef __attribute__((ext_vector_type(4))) float    v4f;
typedef __attribute__((ext_vector_type(8))) float    v8f;
typedef __attribute__((ext_vector_type(4))) unsigned u32x4;
typedef __attribute__((ext_vector_type(8))) int      i32x8;
typedef __attribute__((ext_vector_type(4))) int      i32x4;

#define N_NODES 50000
#define N_EDGES 1600000
#define IN_DIM  128
#define OUT_DIM 64
#define HEADS   4
#define OUTC    (HEADS * OUT_DIM)   // 256

// ---------- helpers: order-preserving float<->uint key for atomic max ----------
__device__ __forceinline__ unsigned f32_key(float x) {
    int i = __float_as_int(x);
    return (i < 0) ? ~(unsigned)i : ((unsigned)i | 0x80000000u);
}
__device__ __forceinline__ float key_f32(unsigned u) {
    int i = (u & 0x80000000u) ? (int)(u & 0x7FFFFFFFu) : (int)(~u);
    return __int_as_float(i);
}

// ---------- TDM: issue a tensor_load_to_lds for a 2D tile ----------
// D# per CDNA5 ISA: group0 = {flags, lds_addr, global_addr, type=2},
// group1 = {data_size=4B, tensor_dim0/1, tile_dim0/1, dim0 stride}.
// 6-arg builtin form (clang-23 / therock-10.0): (g0, g1, g2, g3, g4, cpol).
__device__ __forceinline__ void tdm_load_2d(unsigned lds_addr, const void* gptr,
                                            unsigned td0, unsigned td1,
                                            unsigned t0, unsigned t1, unsigned s0) {
    unsigned long long ga = (unsigned long long)(uintptr_t)gptr;
    u32x4 g0;
    g0.x = 0x1u;                                             // count=1 (valid), user mode
    g0.y = lds_addr;                                         // LDS byte address
    g0.z = (unsigned)(ga & 0xFFFFFFFFull);                   // global_addr[31:0]
    g0.w = (unsigned)((ga >> 32) & 0x01FFFFFFull) | (2u << 30); // addr[56:32] | type=2
    i32x8 g1;
    g1[0] = (int)(2u << 16);                                 // data_size=2 (4 bytes)
    g1[1] = (int)((td0 & 0xFFFFu) << 16);                    // tensor_dim0[15:0]
    g1[2] = (int)(((td0 >> 16) & 0xFFFFu) | ((td1 & 0xFFFFu) << 16));
    g1[3] = (int)(((td1 >> 16) & 0xFFFFu) | ((t0 & 0xFFFFu) << 16)); // tile_dim0
    g1[4] = (int)(t1 & 0xFFFFu);                             // tile_dim1 (tile_dim2=0)
    g1[5] = (int)s0;                                         // tensor_dim0_stride[31:0]
    g1[6] = 0;
    g1[7] = 0;
    i32x4 z4 = {0, 0, 0, 0};
    i32x8 z8 = {0, 0, 0, 0, 0, 0, 0, 0};
    __builtin_amdgcn_tensor_load_to_lds(g0, g1, z4, z4, z8, 0);
}

// ---------- K0: zero output + head stats ----------
__global__ void gat_zero(float* __restrict__ out, unsigned* __restrict__ mkey,
                         float* __restrict__ sums, int n) {
    int i = blockIdx.x * blockDim.x + threadIdx.x;
    if (i < n) out[i] = 0.0f;
    if (i < HEADS) { mkey[i] = 0u; sums[i] = 0.0f; }
}

// ---------- K1: h = x @ W  via TDM -> LDS -> V_WMMA_F32_16X16X4_F32 ----------
// Block (128 threads / 4 waves) computes a 16-row x 64-col (one head) strip.
// blockIdx.x = tm*HEADS + head; wave w owns column tile [w*16, w*16+16).
// LDS: x tile 16x128 (8 KB) + W head panel 128x64 (32 KB), DMA'd by the TDM.
__global__ __launch_bounds__(128) void gat_gemm_wmma(const float* __restrict__ x,
                                                     const float* __restrict__ W,
                                                     float* __restrict__ h) {
    __shared__ float sX[16 * IN_DIM];            // 8192 B
    __shared__ float sW[IN_DIM * OUT_DIM];       // 32768 B

    const int tm   = blockIdx.x / HEADS;
    const int head = blockIdx.x % HEADS;
    const int row0 = tm * 16;

    if ((threadIdx.x >> 5) == 0) {               // wave 0 drives the TDM
        // x tile: 16 rows x 128 cols, row stride 128 elements
        tdm_load_2d((unsigned)(uintptr_t)&sX[0], x + (size_t)row0 * IN_DIM,
                    IN_DIM, 16, IN_DIM, 16, IN_DIM);
        // W head panel: 8192 contiguous elements (1D tile)
        tdm_load_2d((unsigned)(uintptr_t)&sW[0], W + (size_t)head * IN_DIM * OUT_DIM,
                    IN_DIM * OUT_DIM, 1, IN_DIM * OUT_DIM, 0, IN_DIM * OUT_DIM);
        __builtin_amdgcn_s_wait_tensorcnt(0);
    }
    __syncthreads();

    const int wv    = threadIdx.x >> 5;          // column tile within head
    const int lane  = threadIdx.x & 31;
    const int ob    = wv * 16;
    const int arow  = lane & 15;                 // A: row per lane
    const int ncol  = ob + (lane & 15);          // B: col per lane
    const int khalf = (lane >> 4) << 1;          // lanes 16-31 hold K+2,K+3

    v8f c = {};
#pragma unroll
    for (int k = 0; k < IN_DIM; k += 4) {
        v2f a = *(const v2f*)&sX[arow * IN_DIM + k + khalf];   // ds_load_b64
        v2f b;
        b.x = sW[(k + khalf)     * OUT_DIM + ncol];
        b.y = sW[(k + khalf + 1) * OUT_DIM + ncol];
        c = __builtin_amdgcn_wmma_f32_16x16x4_f32(false, a, false, b,
                                                  (short)0, c, false, false);
    }
    // D layout: VGPR v, lane L -> row = row0 + v + 8*(L>=16), col = head*64+ob+(L%16)
    const int rbase = row0 + ((lane >> 4) << 3);
    const int cc    = head * OUT_DIM + ob + (lane & 15);
#pragma unroll
    for (int v = 0; v < 8; ++v)
        h[(size_t)(rbase + v) * OUTC + cc] = c[v];
}

// ---------- K2: per-edge attention logits, wave-per-edge ----------
__global__ __launch_bounds__(256) void gat_edge_scores(const long long* __restrict__ ei,
                                                       const float* __restrict__ h,
                                                       const float* __restrict__ a,
                                                       float* __restrict__ e) {
    const int edge = blockIdx.x * (blockDim.x >> 5) + (threadIdx.x >> 5);
    if (edge >= N_EDGES) return;               // uniform per wave
    const int lane = threadIdx.x & 31;
    const long long s = ei[edge];
    const long long d = ei[N_EDGES + edge];
    const int base = lane << 3;                // 8 floats per lane
    const int head = lane >> 3;

    const v4f* hs = (const v4f*)(h + (size_t)s * OUTC + base);
    const v4f* hd = (const v4f*)(h + (size_t)d * OUTC + base);
    const v4f* as = (const v4f*)(a + head * (2 * OUT_DIM) + ((lane & 7) << 3));
    const v4f* ad = (const v4f*)(a + head * (2 * OUT_DIM) + OUT_DIM + ((lane & 7) << 3));

    float acc = 0.0f;
#pragma unroll
    for (int q = 0; q < 2; ++q) {
        v4f vs = hs[q], vd = hd[q], cs = as[q], cd = ad[q];
        acc += vs.x * cs.x + vs.y * cs.y + vs.z * cs.z + vs.w * cs.w;
        acc += vd.x * cd.x + vd.y * cd.y + vd.z * cd.z + vd.w * cd.w;
    }
    acc += __shfl_xor(acc, 1, 32);
    acc += __shfl_xor(acc, 2, 32);
    acc += __shfl_xor(acc, 4, 32);
    if ((lane & 7) == 0) {
        float v = (acc > 0.0f) ? acc : 0.01f * acc;   // leaky_relu(0.01)
        e[(size_t)edge * HEADS + head] = v;
    }
}

// ---------- K3: global per-head max over all edges ----------
__global__ __launch_bounds__(256) void gat_head_max(const float* __restrict__ e,
                                                    unsigned* __restrict__ mkey) {
    __shared__ unsigned sm[HEADS];
    if (threadIdx.x < HEADS) sm[threadIdx.x] = 0u;
    __syncthreads();
    unsigned lm[HEADS] = {0u, 0u, 0u, 0u};
    const long long stride = (long long)gridDim.x * blockDim.x;
    for (long long i = (long long)blockIdx.x * blockDim.x + threadIdx.x;
         i < N_EDGES; i += stride) {
#pragma unroll
        for (int hd = 0; hd < HEADS; ++hd) {
            unsigned k = f32_key(e[i * HEADS + hd]);
            lm[hd] = (k > lm[hd]) ? k : lm[hd];
        }
    }
#pragma unroll
    for (int hd = 0; hd < HEADS; ++hd) atomicMax(&sm[hd], lm[hd]);
    __syncthreads();
    if (threadIdx.x < HEADS) atomicMax(&mkey[threadIdx.x], sm[threadIdx.x]);
}

// ---------- K4: global per-head sum of exp(e - max) ----------
__global__ __launch_bounds__(256) void gat_head_sum(const float* __restrict__ e,
                                                    const unsigned* __restrict__ mkey,
                                                    float* __restrict__ sums) {
    __shared__ float ss[HEADS];
    if (threadIdx.x < HEADS) ss[threadIdx.x] = 0.0f;
    __syncthreads();
    float m[HEADS];
#pragma unroll
    for (int hd = 0; hd < HEADS; ++hd) m[hd] = key_f32(mkey[hd]);
    float ls[HEADS] = {0.f, 0.f, 0.f, 0.f};
    const long long stride = (long long)gridDim.x * blockDim.x;
    for (long long i = (long long)blockIdx.x * blockDim.x + threadIdx.x;
         i < N_EDGES; i += stride) {
#pragma unroll
        for (int hd = 0; hd < HEADS; ++hd)
            ls[hd] += __expf(e[i * HEADS + hd] - m[hd]);
    }
#pragma unroll
    for (int hd = 0; hd < HEADS; ++hd) atomicAdd(&ss[hd], ls[hd]);
    __syncthreads();
    if (threadIdx.x < HEADS) atomicAdd(&sums[threadIdx.x], ss[threadIdx.x]);
}

// ---------- K5: alpha-weighted scatter (segment sum) ----------
__global__ __launch_bounds__(256) void gat_scatter(const long long* __restrict__ ei,
                                                   const float* __restrict__ h,
                                                   const float* __restrict__ e,
                                                   const unsigned* __restrict__ mkey,
                                                   const float* __restrict__ sums,
                                                   float* __restrict__ out) {
    const int edge = blockIdx.x * (blockDim.x >> 5) + (threadIdx.x >> 5);
    if (edge >= N_EDGES) return;               // uniform per wave
    const int lane = threadIdx.x & 31;
    const long long s = ei[edge];
    const long long d = ei[N_EDGES + edge];
    const int head = lane >> 3;
    const int base = lane << 3;

    const float m = key_f32(mkey[head]);
    const float alpha = __expf(e[(size_t)edge * HEADS + head] - m) / sums[head];

    const float* hs = h + (size_t)s * OUTC + base;
    float* o = out + (size_t)d * OUTC + base;
#pragma unroll
    for (int j = 0; j < 8; ++j)
        atomicAdd(&o[j], hs[j] * alpha);       // GLOBAL_ATOMIC_ADD_F32
}

extern "C" void kernel_launch(void* const* d_in, const int* in_sizes, int n_in,
                              void* d_out, int out_size, void* d_ws, size_t ws_size,
                              hipStream_t stream) {
    const float*     x  = (const float*)d_in[0];                 // (50000,128)
    const long long* ei = (const long long*)d_in[1];             // (2,1.6M) int64
    const float*     W  = (const float*)d_in[2];                 // (4,128,64)
    const float*     a  = (const float*)d_in[3];                 // (4,128)
    float* out = (float*)d_out;                                  // (50000,256)

    // workspace carve-up
    float*    h    = (float*)d_ws;                               // 12.8M floats
    float*    e    = h + (size_t)N_NODES * OUTC;                 // 6.4M floats
    unsigned* mkey = (unsigned*)(e + (size_t)N_EDGES * HEADS);   // 4
    float*    sums = (float*)(mkey + HEADS);                     // 4

    const int nOut = N_NODES * OUTC;                             // 12,800,000
    gat_zero<<<(nOut + 255) / 256, 256, 0, stream>>>(out, mkey, sums, nOut);

    // GEMM: 3125 row tiles x 4 heads, 4 waves/block (one 16x16 tile each)
    gat_gemm_wmma<<<(N_NODES / 16) * HEADS, 128, 0, stream>>>(x, W, h);

    // edge scores: 1.6M edges, 8 edges/block (wave-per-edge)
    gat_edge_scores<<<N_EDGES / 8, 256, 0, stream>>>(ei, h, a, e);

    gat_head_max<<<1024, 256, 0, stream>>>(e, mkey);
    gat_head_sum<<<1024, 256, 0, stream>>>(e, mkey, sums);

    gat_scatter<<<N_EDGES / 8, 256, 0, stream>>>(ei, h, e, mkey, sums, out);
}